// EfficientTensorTrain3D_32152125178389
// MI455X (gfx1250) — compile-verified
//
#include <hip/hip_runtime.h>
#include <math.h>

typedef float v8f __attribute__((ext_vector_type(8)));
typedef float v2f __attribute__((ext_vector_type(2)));

// Workspace layout (floats)
#define WS_ROWD   0        // 128*64   raw sums over (h,w)  -> normalized A1
#define WS_ROWH   8192     // 128*64   raw sums over (d,w)  -> normalized A2
#define WS_ROWW   16384    // 128*64   raw sums over (d,h)  -> normalized A3
#define WS_CHSUM  24576    // 16       per-channel sum(x)
#define WS_CHSQ   24592    // 16       per-channel sum(x^2)
#define WS_B2     24608    // 64*32    B2[h][r] = mean_s core2[r,h,s]
#define WS_ZERON  24608

__global__ void zero_ws_kernel(float* __restrict__ ws, int n) {
    int i = blockIdx.x * blockDim.x + threadIdx.x;
    if (i < n) ws[i] = 0.0f;
}

// Grid: 2048 blocks (128 slabs x 16 d-chunks), 256 threads (8 waves).
// Each block handles 4 d-values x 64 h-rows of one (b,c) slab = 64KB.
// b128 loads: 16 lanes cover one 64-float W-row, so each wave covers TWO rows
// per load; row sums need only a 16-lane (half-wave) shuffle reduction.
__global__ __launch_bounds__(256) void reduce_x_kernel(const float* __restrict__ x,
                                                       float* __restrict__ ws) {
    __shared__ float sD[4];
    __shared__ float sH[64];
    __shared__ float sW[64];
    __shared__ float sSq;

    const int tid  = threadIdx.x;
    const int lane = tid & 31;
    const int wave = tid >> 5;
    const int half = lane >> 4;       // which row of the pair this lane covers
    const int l16  = lane & 15;
    const int bid  = blockIdx.x;
    const int bc    = bid >> 4;
    const int chunk = bid & 15;

    if (tid < 64) { sH[tid] = 0.0f; sW[tid] = 0.0f; }
    if (tid < 4)  sD[tid] = 0.0f;
    if (tid == 64) sSq = 0.0f;
    __syncthreads();

    const float* base = x + (size_t)bc * 262144u + (size_t)chunk * 16384u;

    float accW0 = 0.0f, accW1 = 0.0f, accW2 = 0.0f, accW3 = 0.0f, accSq = 0.0f;
    // 256 rows = 128 row-pairs; each wave takes 16 pairs (32 rows).
    for (int p = wave; p < 128; p += 8) {
        const int r = 2 * p + half;                        // this lane's row
        const float4 v = *reinterpret_cast<const float4*>(base + r * 64 + l16 * 4);
        accW0 += v.x; accW1 += v.y; accW2 += v.z; accW3 += v.w;
        accSq += v.x * v.x + v.y * v.y + v.z * v.z + v.w * v.w;
        float rs = (v.x + v.y) + (v.z + v.w);              // row partial
        #pragma unroll
        for (int off = 8; off > 0; off >>= 1) rs += __shfl_xor(rs, off, 32);
        if (l16 == 0) {                                    // lane 0 -> row 2p, lane 16 -> row 2p+1
            atomicAdd(&sD[r >> 6], rs);
            atomicAdd(&sH[r & 63], rs);
        }
    }
    // fold per-lane W accumulators (lane covers w = (lane&15)*4 + 0..3; both halves add)
    atomicAdd(&sW[l16 * 4 + 0], accW0);
    atomicAdd(&sW[l16 * 4 + 1], accW1);
    atomicAdd(&sW[l16 * 4 + 2], accW2);
    atomicAdd(&sW[l16 * 4 + 3], accW3);
    #pragma unroll
    for (int off = 16; off > 0; off >>= 1) accSq += __shfl_xor(accSq, off, 32);
    if (lane == 0) atomicAdd(&sSq, accSq);
    __syncthreads();

    if (tid < 64) {
        atomicAdd(&ws[WS_ROWW + bc * 64 + tid], sW[tid]);
        atomicAdd(&ws[WS_ROWH + bc * 64 + tid], sH[tid]);
    }
    if (tid < 4)
        atomicAdd(&ws[WS_ROWD + bc * 64 + chunk * 4 + tid], sD[tid]);
    if (tid == 64) {
        float t = sD[0] + sD[1] + sD[2] + sD[3];
        atomicAdd(&ws[WS_CHSUM + (bc & 15)], t);
    }
    if (tid == 65)
        atomicAdd(&ws[WS_CHSQ + (bc & 15)], sSq);
}

// 1 block, 256 threads: channel stats -> affine; normalize summaries in place; B2 pre-reduction.
__global__ __launch_bounds__(256) void prep_kernel(float* __restrict__ ws,
                                                   const float* __restrict__ gamma,
                                                   const float* __restrict__ beta,
                                                   const float* __restrict__ core2) {
    __shared__ float sA[16];
    __shared__ float sOff[16];
    const int tid = threadIdx.x;
    if (tid < 16) {
        const float N = 2097152.0f;                    // 8 * 64^3
        const float s = ws[WS_CHSUM + tid];
        const float q = ws[WS_CHSQ + tid];
        const float mean = s / N;
        const float var  = q / N - mean * mean;
        const float a = gamma[tid] * rsqrtf(var + 1e-5f);
        sA[tid]   = a;
        sOff[tid] = beta[tid] - mean * a;
    }
    __syncthreads();

    // raw axis-sums -> normalized axis-means (all divisors are 4096)
    for (int i = tid; i < 24576; i += 256) {
        const int c = (i >> 6) & 15;                   // bc = (i & 8191) >> 6; c = bc & 15
        ws[i] = sA[c] * (ws[i] * (1.0f / 4096.0f)) + sOff[c];
    }
    // B2[h*32 + r] = mean over s of core2[r, h, s]
    for (int i = tid; i < 2048; i += 256) {
        const int h = i >> 5;
        const int r = i & 31;
        const float* p = core2 + (r * 64 + h) * 32;
        float s = 0.0f;
        #pragma unroll 8
        for (int j = 0; j < 32; ++j) s += p[j];
        ws[WS_B2 + i] = s * (1.0f / 32.0f);
    }
}

// Grid: 8 blocks x 32 threads (one full wave each -> EXEC all ones for WMMA).
// Each wave computes one 16-row M-tile of all three 128x64x32 fp32 GEMMs with
// V_WMMA_F32_16X16X4_F32, then fuses exact-erf GELU and the store.
__global__ __launch_bounds__(32) void gemm_gelu_kernel(const float* __restrict__ ws,
                                                       const float* __restrict__ core1,
                                                       const float* __restrict__ core3,
                                                       float* __restrict__ out) {
    const int lane = threadIdx.x;
    const int m0   = blockIdx.x * 16;
    const int lhi  = lane >> 4;       // 0: lanes 0-15, 1: lanes 16-31
    const int llo  = lane & 15;

    for (int g = 0; g < 3; ++g) {
        const float* A = ws + g * 8192;                       // 128 x 64, row-major
        const float* B = (g == 0) ? core1 : (ws + WS_B2);     // 64 x 32, row-major (g<2)
        for (int nt = 0; nt < 2; ++nt) {
            const int n0 = nt * 16;
            v8f acc = {};
            for (int k0 = 0; k0 < 64; k0 += 4) {
                const int kk = k0 + lhi * 2;                  // A/B K-split per lane half
                v2f a, b;
                const float* ap = A + (m0 + llo) * 64 + kk;
                a.x = ap[0];
                a.y = ap[1];
                const int n = n0 + llo;
                if (g < 2) {
                    b.x = B[kk * 32 + n];
                    b.y = B[(kk + 1) * 32 + n];
                } else {
                    // B3[k=w][n=r] = core3[r*64 + w]  (core3 is (32,64,1))
                    const float* bp = core3 + n * 64 + kk;
                    b.x = bp[0];
                    b.y = bp[1];
                }
                acc = __builtin_amdgcn_wmma_f32_16x16x4_f32(
                    /*neg_a=*/false, a, /*neg_b=*/false, b,
                    /*c_mod=*/(short)0, acc, /*reuse_a=*/false, /*reuse_b=*/false);
            }
            #pragma unroll
            for (int j = 0; j < 8; ++j) {
                const int m = m0 + j + lhi * 8;               // D layout: M=j / M=8+j per lane half
                const int n = n0 + llo;
                const float v = acc[j];
                const float ge = 0.5f * v * (1.0f + erff(v * 0.70710678118654752440f));
                out[m * 96 + g * 32 + n] = ge;
            }
        }
    }
}

extern "C" void kernel_launch(void* const* d_in, const int* in_sizes, int n_in,
                              void* d_out, int out_size, void* d_ws, size_t ws_size,
                              hipStream_t stream) {
    const float* x     = (const float*)d_in[0];
    const float* gamma = (const float*)d_in[1];
    const float* beta  = (const float*)d_in[2];
    const float* core1 = (const float*)d_in[3];
    const float* core2 = (const float*)d_in[4];
    const float* core3 = (const float*)d_in[5];
    float* ws  = (float*)d_ws;
    float* out = (float*)d_out;

    zero_ws_kernel<<<(WS_ZERON + 255) / 256, 256, 0, stream>>>(ws, WS_ZERON);
    reduce_x_kernel<<<2048, 256, 0, stream>>>(x, ws);
    prep_kernel<<<1, 256, 0, stream>>>(ws, gamma, beta, core2);
    gemm_gelu_kernel<<<8, 32, 0, stream>>>(ws, core1, core3, out);
}